// SemanticLevelContext_3D_12_31756988187037
// MI455X (gfx1250) — compile-verified
//
#include <hip/hip_runtime.h>

typedef float v2f __attribute__((ext_vector_type(2)));
typedef float v8f __attribute__((ext_vector_type(8)));

#define NB 2
#define NC 64
#define NK 8

// Async global->LDS path (CDNA5): use builtins when the toolchain declares them.
#if defined(__gfx1250__) && __has_builtin(__builtin_amdgcn_global_load_async_to_lds_b32) && \
    __has_builtin(__builtin_amdgcn_s_wait_asynccnt)
#define USE_ASYNC_LDS 1
typedef __attribute__((address_space(1))) int as1_int;  // printed as '__device__ int*' by clang
typedef __attribute__((address_space(3))) int as3_int;  // printed as '__shared__ int*' by clang
#else
#define USE_ASYNC_LDS 0
#endif

// ws layout (32-bit words): [0,16) mxEnc(u32)  [16,32) den(f32)  [32,1056) segfeats(f32 [16][64])

__device__ __forceinline__ unsigned enc_f32(float f) {
  unsigned u = __float_as_uint(f);
  return (u & 0x80000000u) ? ~u : (u | 0x80000000u);
}
__device__ __forceinline__ float dec_f32(unsigned e) {
  return (e & 0x80000000u) ? __uint_as_float(e & 0x7fffffffu) : __uint_as_float(~e);
}

// ---------------- kernel 0: zero the stats region ----------------
__global__ void k_init(unsigned* ws) {
  for (int i = threadIdx.x; i < 16 + 16 + NK * NC * NB; i += blockDim.x) ws[i] = 0u;
}

// argmax over 8 classes for 4 consecutive voxels (float4 loads)
struct AM4 { float best[4]; int a[4]; };
__device__ __forceinline__ AM4 argmax8x4(const float* __restrict__ pb, int Nv, int v4) {
  AM4 r;
  float4 p = *(const float4*)(pb + v4);
  r.best[0] = p.x; r.best[1] = p.y; r.best[2] = p.z; r.best[3] = p.w;
  r.a[0] = r.a[1] = r.a[2] = r.a[3] = 0;
#pragma unroll
  for (int k = 1; k < NK; ++k) {
    float4 q = *(const float4*)(pb + (size_t)k * Nv + v4);
    if (q.x > r.best[0]) { r.best[0] = q.x; r.a[0] = k; }
    if (q.y > r.best[1]) { r.best[1] = q.y; r.a[1] = k; }
    if (q.z > r.best[2]) { r.best[2] = q.z; r.a[2] = k; }
    if (q.w > r.best[3]) { r.best[3] = q.w; r.a[3] = k; }
  }
  return r;
}

// ---------------- kernel 1: per-segment max of own-class score ----------------
__global__ __launch_bounds__(256) void k_segmax(const float* __restrict__ preds,
                                                unsigned* __restrict__ mxEnc, int Nv) {
  __shared__ unsigned lmx[NK];
  const int b = blockIdx.y;
  if (threadIdx.x < NK) lmx[threadIdx.x] = 0u;
  __syncthreads();
  const float* pb = preds + (size_t)b * NK * Nv;
  const int stride = gridDim.x * blockDim.x * 4;
  for (int v4 = (blockIdx.x * blockDim.x + threadIdx.x) * 4; v4 < Nv; v4 += stride) {
    AM4 r = argmax8x4(pb, Nv, v4);
#pragma unroll
    for (int j = 0; j < 4; ++j) atomicMax(&lmx[r.a[j]], enc_f32(r.best[j]));
  }
  __syncthreads();
  if (threadIdx.x < NK) atomicMax(&mxEnc[b * NK + threadIdx.x], lmx[threadIdx.x]);
}

// ---------------- kernel 2: per-segment softmax denominator ----------------
__global__ __launch_bounds__(256) void k_segden(const float* __restrict__ preds,
                                                const unsigned* __restrict__ mxEnc,
                                                float* __restrict__ den, int Nv) {
  __shared__ float ldn[NK];
  __shared__ float lmx[NK];
  const int b = blockIdx.y;
  if (threadIdx.x < NK) {
    ldn[threadIdx.x] = 0.f;
    lmx[threadIdx.x] = dec_f32(mxEnc[b * NK + threadIdx.x]);
  }
  __syncthreads();
  const float* pb = preds + (size_t)b * NK * Nv;
  const int stride = gridDim.x * blockDim.x * 4;
  for (int v4 = (blockIdx.x * blockDim.x + threadIdx.x) * 4; v4 < Nv; v4 += stride) {
    AM4 r = argmax8x4(pb, Nv, v4);
#pragma unroll
    for (int j = 0; j < 4; ++j) atomicAdd(&ldn[r.a[j]], __expf(r.best[j] - lmx[r.a[j]]));
  }
  __syncthreads();
  if (threadIdx.x < NK) atomicAdd(&den[b * NK + threadIdx.x], ldn[threadIdx.x]);
}

// ---------------- kernel 3: WMMA softmax-weighted feature reduce ----------------
// segfeats[16,64] = W[16,Nv] x F[Nv,64] done as v_wmma_f32_16x16x4_f32 tiles.
// Launch: grid(512, NB), block 128 (4 waves). Nv/32 chunks divide evenly -> uniform loops.
__global__ __launch_bounds__(128) void k_segfeats(const float* __restrict__ x,
                                                  const float* __restrict__ preds,
                                                  const unsigned* __restrict__ mxEnc,
                                                  const float* __restrict__ den,
                                                  float* __restrict__ segfeats, int Nv) {
  __shared__ float smem[4][NC * 33];   // per-wave slab: 64 channels x 32 voxels, padded
  __shared__ float segAcc[NK * NC];    // block-level partial segfeats
  const int b = blockIdx.y;
  const int tid = threadIdx.x;
  const int wid = tid >> 5;
  const int lane = tid & 31;
  const int half = lane >> 4;          // K-half of the wave per WMMA layout
  const int m = lane & 15;             // matrix row held by this lane

  for (int i = tid; i < NK * NC; i += blockDim.x) segAcc[i] = 0.f;
  __syncthreads();

  float mxr[NK], denr[NK];
#pragma unroll
  for (int k = 0; k < NK; ++k) {
    mxr[k] = dec_f32(mxEnc[b * NK + k]);
    denr[k] = den[b * NK + k];
  }

  v8f acc[4];
#pragma unroll
  for (int t = 0; t < 4; ++t) acc[t] = (v8f)0.0f;

  const float* pb = preds + (size_t)b * NK * Nv;
  const float* xb = x + (size_t)b * NC * Nv;
  float* sm = smem[wid];

  const int nch = Nv >> 5;             // 32 voxels per chunk
  const int wstride = gridDim.x * 4;
  for (int ch = blockIdx.x * 4 + wid; ch < nch; ch += wstride) {
    const int v = (ch << 5) + lane;

    // stage 64 channels x 32 voxels of x into LDS (coalesced), padded rows.
    // On CDNA5 route through the async global->LDS engine (ASYNCcnt) so the
    // copy overlaps the argmax/softmax math below; else classic ld/st staging.
#pragma unroll 8
    for (int c = 0; c < NC; ++c) {
      const float* gp = xb + (size_t)c * Nv + v;
      float* lp = &sm[c * 33 + lane];
#if USE_ASYNC_LDS
      __builtin_amdgcn_global_load_async_to_lds_b32((as1_int*)gp, (as3_int*)lp, 0, 0);
#else
      *lp = *gp;
#endif
    }

    // per-lane voxel: argmax class, softmax weight (select segment stats branchlessly)
    float best = pb[v];
    int a = 0;
    float smx = mxr[0], sden = denr[0];
#pragma unroll
    for (int k = 1; k < NK; ++k) {
      float p = pb[(size_t)k * Nv + v];
      bool g = p > best;
      best = g ? p : best;
      a = g ? k : a;
      smx = g ? mxr[k] : smx;
      sden = g ? denr[k] : sden;
    }
    const float wgt = __expf(best - smx) / sden;

#if USE_ASYNC_LDS
    __builtin_amdgcn_s_wait_asynccnt(0);
    asm volatile("" ::: "memory");     // keep LDS reads below the wait
#endif
    __syncthreads();                   // uniform trip counts across the block

    // 8 groups of 4 voxels (K dim), 4 channel tiles of 16 (N dim)
#pragma unroll
    for (int g = 0; g < 8; ++g) {
      const int s0 = g * 4 + half * 2;
      const int a0 = __shfl(a, s0, 32);
      const float w0 = __shfl(wgt, s0, 32);
      const int a1 = __shfl(a, s0 + 1, 32);
      const float w1 = __shfl(wgt, s0 + 1, 32);
      v2f A;
      A.x = (a0 == m) ? w0 : 0.f;
      A.y = (a1 == m) ? w1 : 0.f;
#pragma unroll
      for (int t = 0; t < 4; ++t) {
        const float* bp = &sm[(t * 16 + m) * 33 + g * 4 + half * 2];
        v2f Bm;
        Bm.x = bp[0];
        Bm.y = bp[1];
        acc[t] = __builtin_amdgcn_wmma_f32_16x16x4_f32(
            false, A, false, Bm, (short)0, acc[t], false, false);
      }
    }
    __syncthreads();
  }

  // D layout: lanes 0-15 hold rows M=0..7 (classes) in VGPRs 0..7, col N = lane
  if (lane < 16) {
#pragma unroll
    for (int t = 0; t < 4; ++t)
#pragma unroll
      for (int r = 0; r < NK; ++r) atomicAdd(&segAcc[r * NC + t * 16 + lane], acc[t][r]);
  }
  __syncthreads();
  for (int i = tid; i < NK * NC; i += blockDim.x)
    atomicAdd(&segfeats[b * NK * NC + i], segAcc[i]);
}

// ---------------- kernel 4: out = x * segfeats[seg, c] ----------------
__global__ __launch_bounds__(256) void k_apply(const float* __restrict__ x,
                                               const float* __restrict__ preds,
                                               const float* __restrict__ segfeats,
                                               float* __restrict__ out, int Nv) {
  __shared__ float sf[NK * 65];  // padded to avoid bank conflicts on per-lane class rows
  const int b = blockIdx.y;
  for (int i = threadIdx.x; i < NK * NC; i += blockDim.x)
    sf[(i >> 6) * 65 + (i & 63)] = segfeats[b * NK * NC + i];
  __syncthreads();
  const float* pb = preds + (size_t)b * NK * Nv;
  const float* xb = x + (size_t)b * NC * Nv;
  float* ob = out + (size_t)b * NC * Nv;
  const int stride = gridDim.x * blockDim.x * 4;
  for (int v4 = (blockIdx.x * blockDim.x + threadIdx.x) * 4; v4 < Nv; v4 += stride) {
    AM4 r = argmax8x4(pb, Nv, v4);
    const int o0 = r.a[0] * 65, o1 = r.a[1] * 65, o2 = r.a[2] * 65, o3 = r.a[3] * 65;
#pragma unroll 8
    for (int c = 0; c < NC; ++c) {
      float4 xv = *(const float4*)(xb + (size_t)c * Nv + v4);
      float4 ov;
      ov.x = xv.x * sf[o0 + c];
      ov.y = xv.y * sf[o1 + c];
      ov.z = xv.z * sf[o2 + c];
      ov.w = xv.w * sf[o3 + c];
      *(float4*)(ob + (size_t)c * Nv + v4) = ov;
    }
  }
}

extern "C" void kernel_launch(void* const* d_in, const int* in_sizes, int n_in,
                              void* d_out, int out_size, void* d_ws, size_t ws_size,
                              hipStream_t stream) {
  (void)n_in; (void)out_size; (void)ws_size;
  const float* x = (const float*)d_in[0];
  const float* preds = (const float*)d_in[1];
  float* out = (float*)d_out;

  const int Nv = in_sizes[1] / (NB * NK);  // 64*64*64 = 262144

  unsigned* mxEnc = (unsigned*)d_ws;
  float* den = (float*)d_ws + 16;
  float* segfeats = (float*)d_ws + 32;

  k_init<<<dim3(1), dim3(256), 0, stream>>>((unsigned*)d_ws);
  k_segmax<<<dim3(128, NB), dim3(256), 0, stream>>>(preds, mxEnc, Nv);
  k_segden<<<dim3(128, NB), dim3(256), 0, stream>>>(preds, mxEnc, den, Nv);
  k_segfeats<<<dim3(512, NB), dim3(128), 0, stream>>>(x, preds, mxEnc, den, segfeats, Nv);
  k_apply<<<dim3(256, NB), dim3(256), 0, stream>>>(x, preds, segfeats, out, Nv);
}